// Attention_agent_hv_50525995270752
// MI455X (gfx1250) — compile-verified
//
#include <hip/hip_runtime.h>
#include <hip/hip_bf16.h>

// Agent-attention block for MI455X (gfx1250), fp32 end-to-end using
// V_WMMA_F32_16X16X4_F32 (wave32). Problem is HBM-bandwidth bound
// (~34 GFLOP vs ~500MB traffic @ 23.3 TB/s), so fp32 WMMA keeps full
// accuracy with no roofline penalty vs bf16. Global->LDS staging uses
// CDNA5 async-to-LDS ops (ASYNCcnt) with double buffering in the GEMM.

typedef __attribute__((ext_vector_type(2))) float v2f;
typedef __attribute__((ext_vector_type(8))) float v8f;

#define WMMA_F32(a, b, c) \
  __builtin_amdgcn_wmma_f32_16x16x4_f32(false, (a), false, (b), (short)0, (c), false, false)

static constexpr float kEps   = 1e-5f;
static constexpr float kScale = 0.17677669529663687f;  // 32^-0.5

// ---- CDNA5 async global->LDS copies (inline asm; portable across toolchains)
__device__ __forceinline__ void async_b32(float* lds, const float* g) {
  uint32_t l = (uint32_t)(uintptr_t)lds;   // generic LDS ptr truncates to DS addr
  asm volatile("global_load_async_to_lds_b32 %0, %1, off"
               :: "v"(l), "v"(g) : "memory");
}
__device__ __forceinline__ void async_b128(float* lds, const float* g) {
  uint32_t l = (uint32_t)(uintptr_t)lds;
  asm volatile("global_load_async_to_lds_b128 %0, %1, off"
               :: "v"(l), "v"(g) : "memory");
}
__device__ __forceinline__ void async_wait0() {
  asm volatile("s_wait_asynccnt 0x0" ::: "memory");
}

// ---------------------------------------------------------------------------
// Generic 1x1-conv (GEMM) + BN:  Y[b,m,n] = BN_m( sum_k W[m,k] * X[b,k,n] )
// grid: (4096/64, M/128, B); block 256 (8 waves). Tile 128x64, BK=32,
// double-buffered async staging.
// ---------------------------------------------------------------------------
__global__ __launch_bounds__(256)
void gemm_bn_kernel(const float* __restrict__ Wm, const float* __restrict__ X,
                    const float* __restrict__ bn, float* __restrict__ Y,
                    int M, int K) {
  __shared__ __align__(16) float As[2 * 128 * 40];  // [buf][m_local][k] pitch 40
  __shared__ __align__(16) float Bs[2 * 64 * 40];   // [buf][n_local][k]
  __shared__ float sInv[128], sAdd[128];

  const int t = threadIdx.x;
  const int lane = t & 31, wave = t >> 5;
  const int l15 = lane & 15, lhi = (lane >> 4) & 1;
  const int nBase = blockIdx.x * 64;
  const int mBase = blockIdx.y * 128;
  const int b = blockIdx.z;

  if (t < 128) {
    int m = mBase + t;
    float g = bn[m], be = bn[M + m], mu = bn[2 * M + m], va = bn[3 * M + m];
    float inv = g * rsqrtf(va + kEps);
    sInv[t] = inv;
    sAdd[t] = be - mu * inv;
  }

  const float* Xb = X + (size_t)b * K * 4096 + nBase;
  const int nkt = K >> 5;

  auto stage = [&](int kt, int buf) {
    float* Ad = As + buf * (128 * 40);
    const float* Asrc = Wm + (size_t)mBase * K + kt;
#pragma unroll
    for (int p = 0; p < 4; ++p) {                 // A: 128x32, 4 floats/op
      int idx = (p * 256 + t) * 4;
      int i = idx >> 5, j = idx & 31;
      async_b128(Ad + i * 40 + j, Asrc + (size_t)i * K + j);
    }
    float* Bd = Bs + buf * (64 * 40);
    const float* Bsrc = Xb + (size_t)kt * 4096;
#pragma unroll
    for (int p = 0; p < 8; ++p) {                 // B transposed: [n][k]
      int idx = p * 256 + t;
      int kk = idx >> 6, nn = idx & 63;
      async_b32(Bd + nn * 40 + kk, Bsrc + (size_t)kk * 4096 + nn);
    }
  };

  v8f acc[4] = {};
  stage(0, 0);
  for (int kc = 0; kc < nkt; ++kc) {
    async_wait0();          // my slice of buf kc is in LDS
    __syncthreads();        // everyone's slice is in LDS
    if (kc + 1 < nkt) stage((kc + 1) << 5, (kc + 1) & 1);  // prefetch next tile

    const float* Ab = As + (kc & 1) * (128 * 40);
    const float* Bb = Bs + (kc & 1) * (64 * 40);
    const float* Arow = Ab + (wave * 16 + l15) * 40 + 2 * lhi;
#pragma unroll
    for (int k4 = 0; k4 < 8; ++k4) {
      v2f a = *(const v2f*)(Arow + k4 * 4);
#pragma unroll
      for (int cb = 0; cb < 4; ++cb) {
        v2f bf = *(const v2f*)(Bb + (cb * 16 + l15) * 40 + k4 * 4 + 2 * lhi);
        acc[cb] = WMMA_F32(a, bf, acc[cb]);
      }
    }
    __syncthreads();        // done reading buf kc; safe to overwrite at kc+2
  }

#pragma unroll
  for (int cb = 0; cb < 4; ++cb)
#pragma unroll
    for (int r = 0; r < 8; ++r) {
      int lm = wave * 16 + r + 8 * lhi;           // C/D layout: row r+8*lanehi
      int m = mBase + lm;
      int n = nBase + cb * 16 + l15;
      Y[((size_t)b * M + m) * 4096 + n] = acc[cb][r] * sInv[lm] + sAdd[lm];
    }
}

// ---------------------------------------------------------------------------
// Pool: agent tokens (8x8 block-mean of q) and W-means of k,v.
// grid B*C, block 64.
// ---------------------------------------------------------------------------
__global__ __launch_bounds__(64)
void pool_kernel(const float* __restrict__ qkv, float* __restrict__ a_mat,
                 float* __restrict__ ak, float* __restrict__ av) {
  int gid = blockIdx.x;          // b*256+c
  int b = gid >> 8, c = gid & 255;
  int t = threadIdx.x;
  const float* q = qkv + ((size_t)b * 768 + c) * 4096;
  const float* k = qkv + ((size_t)b * 768 + 256 + c) * 4096;
  const float* v = qkv + ((size_t)b * 768 + 512 + c) * 4096;

  int py = t >> 3, px = t & 7;   // agent = py*8+px
  float s = 0.f;
  for (int yy = 0; yy < 8; ++yy)
    for (int xx = 0; xx < 8; ++xx)
      s += q[(py * 8 + yy) * 64 + px * 8 + xx];
  a_mat[((size_t)(b * 8 + (c >> 5)) * 32 + (c & 31)) * 64 + t] = s * (1.f / 64.f);

  float sk = 0.f, sv = 0.f;
  const float* kr = k + t * 64;
  const float* vr = v + t * 64;
  for (int xx = 0; xx < 64; ++xx) { sk += kr[xx]; sv += vr[xx]; }
  ak[(size_t)gid * 64 + t] = sk * (1.f / 64.f);
  av[(size_t)gid * 64 + t] = sv * (1.f / 64.f);
}

// ---------------------------------------------------------------------------
// SE gates: gate = BN2(cv2 @ BN1(cv1 @ pooled)). grid B, block 256.
// ---------------------------------------------------------------------------
__global__ __launch_bounds__(256)
void gate_kernel(const float* __restrict__ w1, const float* __restrict__ bn1,
                 const float* __restrict__ w2, const float* __restrict__ bn2,
                 const float* __restrict__ ak, const float* __restrict__ av,
                 float* __restrict__ gk, float* __restrict__ gv) {
  __shared__ float t1k[8][64], t1v[8][64];
  int b = blockIdx.x, t = threadIdx.x;
  for (int p = 0; p < 2; ++p) {
    int idx = p * 256 + t;       // 512 outputs (m,y)
    int m = idx >> 6, y = idx & 63;
    float sk = 0.f, sv = 0.f;
    for (int c = 0; c < 256; ++c) {
      float w = w1[m * 256 + c];
      sk += w * ak[((size_t)b * 256 + c) * 64 + y];
      sv += w * av[((size_t)b * 256 + c) * 64 + y];
    }
    float g = bn1[m], be = bn1[8 + m], mu = bn1[16 + m], va = bn1[24 + m];
    float inv = g * rsqrtf(va + kEps), add = be - mu * inv;
    t1k[m][y] = sk * inv + add;
    t1v[m][y] = sv * inv + add;
  }
  __syncthreads();
  int c = t;
  float g = bn2[c], be = bn2[256 + c], mu = bn2[512 + c], va = bn2[768 + c];
  float inv = g * rsqrtf(va + kEps), add = be - mu * inv;
  for (int y = 0; y < 64; ++y) {
    float sk = 0.f, sv = 0.f;
    for (int m = 0; m < 8; ++m) {
      float w = w2[c * 8 + m];
      sk += w * t1k[m][y];
      sv += w * t1v[m][y];
    }
    gk[((size_t)b * 256 + c) * 64 + y] = sk * inv + add;
    gv[((size_t)b * 256 + c) * 64 + y] = sv * inv + add;
  }
}

// Apply gates in place on the k/v sections of qkv. grid (B*C, 16), block 256.
__global__ __launch_bounds__(256)
void apply_gate_kernel(float* __restrict__ qkv, const float* __restrict__ gk,
                       const float* __restrict__ gv) {
  int gid = blockIdx.x;
  int b = gid >> 8, c = gid & 255;
  int n = blockIdx.y * 256 + threadIdx.x;
  int y = n >> 6;
  float fk = gk[((size_t)b * 256 + c) * 64 + y];
  float fv = gv[((size_t)b * 256 + c) * 64 + y];
  qkv[((size_t)b * 768 + 256 + c) * 4096 + n] *= fk;
  qkv[((size_t)b * 768 + 512 + c) * 4096 + n] *= fv;
}

// ---------------------------------------------------------------------------
// S1[a,n] = scale * sum_d a[d,a] * kg[d,n].  grid (4096/64, B*NH), block 128.
// ---------------------------------------------------------------------------
__global__ __launch_bounds__(128)
void attn1_kernel(const float* __restrict__ qkv, const float* __restrict__ a_mat,
                  float* __restrict__ S1) {
  __shared__ __align__(16) float aT[64 * 40];  // [agent][d]
  __shared__ __align__(16) float kT[64 * 40];  // [n][d]
  int t = threadIdx.x, lane = t & 31, wave = t >> 5;
  int l15 = lane & 15, lhi = (lane >> 4) & 1;
  int nBase = blockIdx.x * 64;
  int bh = blockIdx.y, b = bh >> 3, h = bh & 7;
  const float* kg = qkv + ((size_t)b * 768 + 256 + h * 32) * 4096;

  for (int p = 0; p < 16; ++p) {
    int idx = p * 128 + t;       // 2048 elems each
    int d = idx >> 6, j = idx & 63;
    async_b32(&aT[j * 40 + d], &a_mat[((size_t)bh * 32 + d) * 64 + j]);
    async_b32(&kT[j * 40 + d], &kg[(size_t)d * 4096 + nBase + j]);
  }
  async_wait0();
  __syncthreads();

  v8f acc[4] = {};
#pragma unroll
  for (int k4 = 0; k4 < 8; ++k4) {
    v2f a = *(const v2f*)(aT + (wave * 16 + l15) * 40 + k4 * 4 + 2 * lhi);
#pragma unroll
    for (int cb = 0; cb < 4; ++cb) {
      v2f bf = *(const v2f*)(kT + (cb * 16 + l15) * 40 + k4 * 4 + 2 * lhi);
      acc[cb] = WMMA_F32(a, bf, acc[cb]);
    }
  }
#pragma unroll
  for (int cb = 0; cb < 4; ++cb)
#pragma unroll
    for (int r = 0; r < 8; ++r) {
      int arow = wave * 16 + r + 8 * lhi;
      S1[((size_t)bh * 64 + arow) * 4096 + nBase + cb * 16 + l15] =
          acc[cb][r] * kScale;
    }
}

// Row softmax over 4096 columns, in place. grid B*NH*AG, block 256.
__global__ __launch_bounds__(256)
void softmax_rows_kernel(float* __restrict__ S) {
  __shared__ float red[256];
  float* p = S + (size_t)blockIdx.x * 4096;
  int t = threadIdx.x;
  float vals[16], mx = -1e30f;
  for (int i = 0; i < 16; ++i) { vals[i] = p[i * 256 + t]; mx = fmaxf(mx, vals[i]); }
  red[t] = mx; __syncthreads();
  for (int s = 128; s > 0; s >>= 1) { if (t < s) red[t] = fmaxf(red[t], red[t + s]); __syncthreads(); }
  mx = red[0]; __syncthreads();
  float sum = 0.f;
  for (int i = 0; i < 16; ++i) { vals[i] = __expf(vals[i] - mx); sum += vals[i]; }
  red[t] = sum; __syncthreads();
  for (int s = 128; s > 0; s >>= 1) { if (t < s) red[t] += red[t + s]; __syncthreads(); }
  float rs = 1.f / red[0];
  for (int i = 0; i < 16; ++i) p[i * 256 + t] = vals[i] * rs;
}

// ---------------------------------------------------------------------------
// attn partials: attnp[ch][bh][a][d] = sum_{n in chunk} P[a,n] * vg[d,n].
// grid (8 chunks, B*NH), block 256 (8 waves: 4 a-strips x 2 d-blocks).
// Fixed chunk partials + ordered reduce => bit-deterministic (no fp atomics).
// ---------------------------------------------------------------------------
__global__ __launch_bounds__(256)
void attn3_kernel(const float* __restrict__ qkv, const float* __restrict__ P,
                  float* __restrict__ attnp) {
  int t = threadIdx.x, lane = t & 31, wave = t >> 5;
  int l15 = lane & 15, lhi = (lane >> 4) & 1;
  int bh = blockIdx.y, b = bh >> 3, h = bh & 7;
  int n0 = blockIdx.x * 512;
  int strip = wave >> 1, cb = wave & 1;
  const float* A = P + ((size_t)bh * 64 + strip * 16 + l15) * 4096 + n0 + 2 * lhi;
  const float* Bv = qkv +
      ((size_t)(b * 768 + 512 + h * 32 + cb * 16 + l15)) * 4096 + n0 + 2 * lhi;

  v8f acc = {};
#pragma unroll 4
  for (int k = 0; k < 512; k += 4) {
    v2f a = *(const v2f*)(A + k);
    v2f bf = *(const v2f*)(Bv + k);
    acc = WMMA_F32(a, bf, acc);
  }
#pragma unroll
  for (int r = 0; r < 8; ++r) {
    int a = strip * 16 + r + 8 * lhi;
    int d = cb * 16 + l15;
    attnp[(((size_t)blockIdx.x * 64 + bh) * 64 + a) * 32 + d] = acc[r];
  }
}

__global__ __launch_bounds__(256)
void reduce8_kernel(const float* __restrict__ attnp, float* __restrict__ attnb) {
  int i = blockIdx.x * 256 + threadIdx.x;   // < 131072
  float s = 0.f;
  for (int ch = 0; ch < 8; ++ch) s += attnp[(size_t)ch * 131072 + i];
  attnb[i] = s;
}

// ---------------------------------------------------------------------------
// Fused: S2 = qT*a (WMMA) -> softmax over 64 agents -> out = attnT * PT (WMMA).
// grid (4096/64, B*NH), block 128.
// ---------------------------------------------------------------------------
__global__ __launch_bounds__(128)
void attn2_kernel(const float* __restrict__ qkv, const float* __restrict__ a_mat,
                  const float* __restrict__ attnb, float* __restrict__ OUT) {
  __shared__ __align__(16) float qT[64 * 40];    // [n][d]
  __shared__ __align__(16) float aT[64 * 40];    // [agent][d]
  __shared__ __align__(16) float atT[32 * 72];   // [d][agent]
  __shared__ __align__(16) float Pl[64 * 72];    // [n][agent]
  int t = threadIdx.x, lane = t & 31, wave = t >> 5;
  int l15 = lane & 15, lhi = (lane >> 4) & 1;
  int nBase = blockIdx.x * 64;
  int bh = blockIdx.y, b = bh >> 3, h = bh & 7;
  const float* q = qkv + ((size_t)b * 768 + h * 32) * 4096;

  for (int p = 0; p < 16; ++p) {
    int idx = p * 128 + t;
    int d = idx >> 6, j = idx & 63;
    async_b32(&qT[j * 40 + d], &q[(size_t)d * 4096 + nBase + j]);
    async_b32(&aT[j * 40 + d], &a_mat[((size_t)bh * 32 + d) * 64 + j]);
    async_b32(&atT[d * 72 + j], &attnb[((size_t)bh * 64 + j) * 32 + d]);
  }
  async_wait0();
  __syncthreads();

  // Phase 1: S2 tile [64n x 64a], K = 32 (d)
  v8f acc[4] = {};
#pragma unroll
  for (int k4 = 0; k4 < 8; ++k4) {
    v2f a = *(const v2f*)(qT + (wave * 16 + l15) * 40 + k4 * 4 + 2 * lhi);
#pragma unroll
    for (int ab = 0; ab < 4; ++ab) {
      v2f bf = *(const v2f*)(aT + (ab * 16 + l15) * 40 + k4 * 4 + 2 * lhi);
      acc[ab] = WMMA_F32(a, bf, acc[ab]);
    }
  }
#pragma unroll
  for (int ab = 0; ab < 4; ++ab)
#pragma unroll
    for (int r = 0; r < 8; ++r)
      Pl[(wave * 16 + r + 8 * lhi) * 72 + ab * 16 + l15] = acc[ab][r] * kScale;
  __syncthreads();

  // Softmax over the 64 agents per n-row
  if (t < 64) {
    float* row = Pl + t * 72;
    float mx = -1e30f;
    for (int i = 0; i < 64; ++i) mx = fmaxf(mx, row[i]);
    float sum = 0.f;
    for (int i = 0; i < 64; ++i) { float e = __expf(row[i] - mx); row[i] = e; sum += e; }
    float rs = 1.f / sum;
    for (int i = 0; i < 64; ++i) row[i] *= rs;
  }
  __syncthreads();

  // Phase 2: out [32d x 64n] = attnT(32x64) x P^T, K = 64 (agents)
  v8f o[2] = {};
#pragma unroll
  for (int s = 0; s < 2; ++s) {
#pragma unroll
    for (int k4 = 0; k4 < 16; ++k4) {
      v2f a = *(const v2f*)(atT + (s * 16 + l15) * 72 + k4 * 4 + 2 * lhi);
      v2f bf = *(const v2f*)(Pl + (wave * 16 + l15) * 72 + k4 * 4 + 2 * lhi);
      o[s] = WMMA_F32(a, bf, o[s]);
    }
  }
#pragma unroll
  for (int s = 0; s < 2; ++s)
#pragma unroll
    for (int r = 0; r < 8; ++r) {
      int d = s * 16 + r + 8 * lhi;
      OUT[((size_t)(b * 256 + h * 32 + d)) * 4096 + nBase + wave * 16 + l15] =
          o[s][r];
    }
}

// Depthwise 3x3 positional encoding on gated v, + BN, accumulated into OUT.
__global__ __launch_bounds__(256)
void pe_kernel(const float* __restrict__ qkv, const float* __restrict__ w_pe,
               const float* __restrict__ bn_pe, float* __restrict__ OUT) {
  int gid = blockIdx.x;          // b*256+c
  int b = gid >> 8, c = gid & 255;
  int t = threadIdx.x;
  const float* v = qkv + ((size_t)b * 768 + 512 + c) * 4096;
  float wl[9];
#pragma unroll
  for (int i = 0; i < 9; ++i) wl[i] = w_pe[c * 9 + i];
  float g = bn_pe[c], be = bn_pe[256 + c], mu = bn_pe[512 + c], va = bn_pe[768 + c];
  float inv = g * rsqrtf(va + kEps), add = be - mu * inv;
  for (int i = 0; i < 16; ++i) {
    int n = i * 256 + t;
    int y = n >> 6, x = n & 63;
    float s = 0.f;
#pragma unroll
    for (int ky = 0; ky < 3; ++ky) {
      int yy = y + ky - 1;
      if (yy < 0 || yy >= 64) continue;
#pragma unroll
      for (int kx = 0; kx < 3; ++kx) {
        int xx = x + kx - 1;
        if (xx < 0 || xx >= 64) continue;
        s += wl[ky * 3 + kx] * v[yy * 64 + xx];
      }
    }
    OUT[((size_t)(b * 256 + c)) * 4096 + n] += s * inv + add;
  }
}

// ---------------------------------------------------------------------------
extern "C" void kernel_launch(void* const* d_in, const int* in_sizes, int n_in,
                              void* d_out, int out_size, void* d_ws, size_t ws_size,
                              hipStream_t stream) {
  const float* x       = (const float*)d_in[0];
  const float* w_qkv   = (const float*)d_in[1];
  const float* bn_qkv  = (const float*)d_in[2];
  const float* w_cv1   = (const float*)d_in[3];
  const float* bn_cv1  = (const float*)d_in[4];
  const float* w_cv2   = (const float*)d_in[5];
  const float* bn_cv2  = (const float*)d_in[6];
  const float* w_proj  = (const float*)d_in[7];
  const float* bn_proj = (const float*)d_in[8];
  const float* w_pe    = (const float*)d_in[9];
  const float* bn_pe   = (const float*)d_in[10];
  float* out = (float*)d_out;

  float* ws    = (float*)d_ws;
  float* qkv   = ws;                  // 8*768*4096       = 25165824 f
  float* S1    = qkv + 25165824;      // 8*8*64*4096      = 16777216 f
  float* OUT   = S1 + 16777216;       // 8*256*4096       = 8388608 f
  float* a_mat = OUT + 8388608;       // 131072 f
  float* ak    = a_mat + 131072;
  float* av    = ak + 131072;
  float* gk    = av + 131072;
  float* gv    = gk + 131072;
  float* attnb = gv + 131072;         // 131072 f
  float* attnp = attnb + 131072;      // 8*131072 = 1048576 f

  // 1) qkv = BN(w_qkv @ x)
  gemm_bn_kernel<<<dim3(64, 6, 8), 256, 0, stream>>>(w_qkv, x, bn_qkv, qkv, 768, 256);
  // 2) pooled means + agent tokens
  pool_kernel<<<2048, 64, 0, stream>>>(qkv, a_mat, ak, av);
  // 3) SE gates
  gate_kernel<<<8, 256, 0, stream>>>(w_cv1, bn_cv1, w_cv2, bn_cv2, ak, av, gk, gv);
  // 4) k,v *= gates (in place)
  apply_gate_kernel<<<dim3(2048, 16), 256, 0, stream>>>(qkv, gk, gv);
  // 5) S1 = scale * a^T k
  attn1_kernel<<<dim3(64, 64), 128, 0, stream>>>(qkv, a_mat, S1);
  // 6) softmax over N
  softmax_rows_kernel<<<4096, 256, 0, stream>>>(S1);
  // 7) attn = P v^T (8 deterministic partials + reduce)
  attn3_kernel<<<dim3(8, 64), 256, 0, stream>>>(qkv, S1, attnp);
  reduce8_kernel<<<512, 256, 0, stream>>>(attnp, attnb);
  // 8) out = softmax(q^T a) attn
  attn2_kernel<<<dim3(64, 64), 128, 0, stream>>>(qkv, a_mat, attnb, OUT);
  // 9) out += BN(depthwise3x3(v))
  pe_kernel<<<2048, 256, 0, stream>>>(qkv, w_pe, bn_pe, OUT);
  // 10) final projection + BN
  gemm_bn_kernel<<<dim3(64, 2, 8), 256, 0, stream>>>(w_proj, OUT, bn_proj, out, 256, 256);
}